// Predicting_Base_83743272337962
// MI455X (gfx1250) — compile-verified
//
#include <hip/hip_runtime.h>
#include <stdint.h>

#define BATCH    32
#define NCAND    200000
#define TOPK     500
#define KPAD     512
#define KEEP_MAX 100
#define CONF_THR 0.05f
#define NMS_THR  0.5f
#define NBINS    4096
#define SPLIT    32
#define CHUNK    6400   // multiple of 256 and of 4 -> float4-aligned chunks; 32*6400 >= 200000

typedef unsigned int       u32;
typedef unsigned long long u64;
typedef u32 u32x4 __attribute__((ext_vector_type(4)));
typedef u32 u32x8 __attribute__((ext_vector_type(8)));

__device__ __forceinline__ int score_bin(float s) {
  int b = (int)(s * (float)NBINS);
  return b < 0 ? 0 : (b > NBINS - 1 ? NBINS - 1 : b);
}
// monotonic bijection float -> u32 (ascending)
__device__ __forceinline__ u32 orderable(float f) {
  u32 u = __float_as_uint(f);
  return (u & 0x80000000u) ? ~u : (u | 0x80000000u);
}
__device__ __forceinline__ float unorderable(u32 u) {
  u = (u & 0x80000000u) ? (u & 0x7FFFFFFFu) : ~u;
  return __uint_as_float(u);
}

// ---- Tensor Data Mover: 1-D tile of `dwords` 4-byte elements, global -> LDS ----
__device__ __forceinline__ void tdm_load_1d(void* lds_dst, const void* gsrc, u32 dwords) {
  u64 gaddr = (u64)(uintptr_t)gsrc;
  u32 lds_addr = (u32)(uintptr_t)lds_dst;   // low 32 bits = LDS byte address
  u32x4 g0;
  g0[0] = 1u;                                              // count=1 valid D#
  g0[1] = lds_addr;                                        // lds_addr[31:0]
  g0[2] = (u32)gaddr;                                      // global_addr[31:0]
  g0[3] = ((u32)(gaddr >> 32) & 0x01FFFFFFu) | (2u << 30); // addr[56:32] | type=2
  u32x8 g1;
  g1[0] = 0x00020000u;                                     // data_size=2 -> 4 bytes
  g1[1] = (dwords & 0xFFFFu) << 16;                        // tensor_dim0[15:0]
  g1[2] = (dwords >> 16) | (1u << 16);                     // dim0 hi | tensor_dim1=1
  g1[3] = (dwords & 0xFFFFu) << 16;                        // tile_dim0
  g1[4] = 0u;                                              // tile_dim1/2 unused
  g1[5] = dwords;                                          // tensor_dim0_stride
  g1[6] = 0u;
  g1[7] = 0u;
  asm volatile("tensor_load_to_lds %0, %1" :: "s"(g0), "s"(g1) : "memory");
}
__device__ __forceinline__ void tdm_wait0() {
#if __has_builtin(__builtin_amdgcn_s_wait_tensorcnt)
  __builtin_amdgcn_s_wait_tensorcnt(0);
#else
  asm volatile("s_wait_tensorcnt 0" ::: "memory");
#endif
}

// ---------------------------------------------------------------- init ----
__global__ void k_init(u32* hist, float* cscore, u32* cidx) {
  int stride = gridDim.x * blockDim.x;
  int tid = blockIdx.x * blockDim.x + threadIdx.x;
  for (int i = tid; i < BATCH * NBINS; i += stride) hist[i] = 0u;
  for (int i = tid; i < BATCH * KPAD; i += stride) {
    cscore[i] = -3.0e38f;       // sorts to the end (descending sort)
    cidx[i]   = 0xFFFFFFFFu;    // sentinel -> invalid
  }
}

// ------------------------------------- histogram (BATCH x SPLIT blocks) ----
// Only pass that touches HBM (25.6 MB); use b128 loads to feed 23.3 TB/s.
__global__ void k_hist(const float* __restrict__ pscores, u32* __restrict__ g_hist) {
  __shared__ u32 lh[NBINS];
  const int b = blockIdx.x, sblk = blockIdx.y;
  for (int i = threadIdx.x; i < NBINS; i += blockDim.x) lh[i] = 0u;
  __syncthreads();
  const float* row = pscores + (size_t)b * NCAND;
  const int lo = sblk * CHUNK;
  const int hiend = (lo + CHUNK < NCAND) ? lo + CHUNK : NCAND;
  for (int i = lo + threadIdx.x * 4; i + 4 <= hiend; i += blockDim.x * 4) {
    __builtin_prefetch(row + i + 8 * blockDim.x * 4, 0, 1);  // global_prefetch_b8
    float4 v = *(const float4*)(row + i);                    // global_load_b128
    atomicAdd(&lh[score_bin(v.x)], 1u);
    atomicAdd(&lh[score_bin(v.y)], 1u);
    atomicAdd(&lh[score_bin(v.z)], 1u);
    atomicAdd(&lh[score_bin(v.w)], 1u);
  }
  __syncthreads();
  for (int i = threadIdx.x; i < NBINS; i += blockDim.x) {
    u32 c = lh[i];
    if (c) atomicAdd(&g_hist[b * NBINS + i], c);   // integer adds: deterministic
  }
}

// -------------------------------------------------------------- cutoff ----
__global__ void k_cut(const u32* __restrict__ g_hist, u32* __restrict__ g_cut) {
  int b = threadIdx.x;
  if (b >= BATCH) return;
  const u32* h = g_hist + b * NBINS;
  u32 cnt = 0; int cb = 0; u32 eq = 0;
  for (int bin = NBINS - 1; bin >= 0; --bin) {
    u32 c = h[bin];
    if (cnt + c >= TOPK || bin == 0) { cb = bin; eq = c; break; }
    cnt += c;
  }
  u32 room = KPAD - cnt;
  u32 take = eq < room ? eq : room;   // extra boundary candidates; sort fixes order
  g_cut[b * 4 + 0] = (u32)cb;
  g_cut[b * 4 + 1] = cnt;             // strictly-above count
  g_cut[b * 4 + 2] = take;
  g_cut[b * 4 + 3] = cnt + take;
}

// ---------------------------- compaction phase A: per-block counts ----
// Re-reads scores (L2-resident after k_hist; 192 MB global L2).
__global__ void k_cnt(const float* __restrict__ pscores, const u32* __restrict__ g_cut,
                      u32* __restrict__ blk_hi, u32* __restrict__ blk_eq) {
  __shared__ u32 t_hi, t_eq;
  const int b = blockIdx.x, sblk = blockIdx.y, tid = threadIdx.x;
  const int cb = (int)g_cut[b * 4 + 0];
  if (tid == 0) { t_hi = 0u; t_eq = 0u; }
  __syncthreads();
  const float* row = pscores + (size_t)b * NCAND;
  const int lo = sblk * CHUNK;
  const int hiend = (lo + CHUNK < NCAND) ? lo + CHUNK : NCAND;
  u32 chi = 0, ceq = 0;
  for (int i = lo + tid * 4; i + 4 <= hiend; i += blockDim.x * 4) {
    float4 v = *(const float4*)(row + i);                    // global_load_b128
    int b0 = score_bin(v.x), b1 = score_bin(v.y);
    int b2 = score_bin(v.z), b3 = score_bin(v.w);
    chi += (b0 > cb) + (b1 > cb) + (b2 > cb) + (b3 > cb);
    ceq += (b0 == cb) + (b1 == cb) + (b2 == cb) + (b3 == cb);
  }
  atomicAdd(&t_hi, chi);
  atomicAdd(&t_eq, ceq);
  __syncthreads();
  if (tid == 0) {
    blk_hi[b * SPLIT + sblk] = t_hi;
    blk_eq[b * SPLIT + sblk] = t_eq;
  }
}

// ---------------------------- compaction phase B: exclusive block scan ----
__global__ void k_scan(const u32* __restrict__ blk_hi, const u32* __restrict__ blk_eq,
                       u32* __restrict__ base_hi, u32* __restrict__ base_eq) {
  int b = threadIdx.x;
  if (b >= BATCH) return;
  u32 rh = 0, re = 0;
  for (int s = 0; s < SPLIT; ++s) {
    base_hi[b * SPLIT + s] = rh;
    base_eq[b * SPLIT + s] = re;
    rh += blk_hi[b * SPLIT + s];
    re += blk_eq[b * SPLIT + s];
  }
}

// ---------------------------- compaction phase C: deterministic place ----
__global__ void k_place(const float* __restrict__ pscores, const u32* __restrict__ g_cut,
                        const u32* __restrict__ base_hi, const u32* __restrict__ base_eq,
                        float* __restrict__ cscore, u32* __restrict__ cidx) {
  __shared__ u32 s_whi[8], s_weq[8];
  __shared__ u32 s_hibase, s_eqbase;
  const int b = blockIdx.x, sblk = blockIdx.y;
  const int tid = threadIdx.x, lane = tid & 31, wid = tid >> 5;
  const u32 cb = g_cut[b * 4 + 0], n_above = g_cut[b * 4 + 1], take = g_cut[b * 4 + 2];
  if (tid == 0) {
    s_hibase = base_hi[b * SPLIT + sblk];
    s_eqbase = base_eq[b * SPLIT + sblk];
  }
  const float* row = pscores + (size_t)b * NCAND;
  float* cs = cscore + b * KPAD;
  u32*   ci = cidx   + b * KPAD;
  const int lo = sblk * CHUNK;
  const int hiend = (lo + CHUNK < NCAND) ? lo + CHUNK : NCAND;
  __syncthreads();
  for (int base = lo; base < hiend; base += blockDim.x) {
    int i = base + tid;
    bool inr = i < hiend;
    float s = 0.0f; int bin = -1;
    if (inr) { s = row[i]; bin = score_bin(s); }
    bool hi = inr && bin > (int)cb;
    bool eq = inr && bin == (int)cb;
    u64 mhi = __ballot(hi);
    u64 meq = __ballot(eq);
    u32 pre_hi = (u32)__popcll(mhi & ((1ull << lane) - 1ull));
    u32 pre_eq = (u32)__popcll(meq & ((1ull << lane) - 1ull));
    if (lane == 0) { s_whi[wid] = (u32)__popcll(mhi); s_weq[wid] = (u32)__popcll(meq); }
    __syncthreads();
    u32 woff_hi = 0, woff_eq = 0, tot_hi = 0, tot_eq = 0;
    for (int w = 0; w < 8; ++w) {
      u32 a = s_whi[w], c = s_weq[w];
      if (w < wid) { woff_hi += a; woff_eq += c; }
      tot_hi += a; tot_eq += c;
    }
    if (hi) {
      u32 p = s_hibase + woff_hi + pre_hi;
      if (p < KPAD) { cs[p] = s; ci[p] = (u32)i; }
    } else if (eq) {
      u32 p = s_eqbase + woff_eq + pre_eq;
      if (p < take) { cs[n_above + p] = s; ci[n_above + p] = (u32)i; }
    }
    __syncthreads();
    if (tid == 0) { s_hibase += tot_hi; s_eqbase += tot_eq; }
    __syncthreads();
  }
}

// ------------------------------------------- bitonic sort + gather ----
__global__ void k_sort(const float* __restrict__ pboxes, const int* __restrict__ plabels,
                       const float* __restrict__ cscore, const u32* __restrict__ cidx,
                       float* __restrict__ scoresk, float* __restrict__ boxesk,
                       int* __restrict__ labelsk) {
  __shared__ u64 key[KPAD];
  const int b = blockIdx.x, tid = threadIdx.x;
  for (int t = tid; t < KPAD; t += blockDim.x) {
    float s  = cscore[b * KPAD + t];
    u32  idx = cidx  [b * KPAD + t];
    key[t] = ((u64)orderable(s) << 32) | (u32)(~idx);  // score desc, idx asc tiebreak
  }
  __syncthreads();
  for (int k = 2; k <= KPAD; k <<= 1)
    for (int j = k >> 1; j > 0; j >>= 1) {
      for (int t = tid; t < KPAD; t += blockDim.x) {
        int ixj = t ^ j;
        if (ixj > t) {
          u64 a = key[t], c = key[ixj];
          bool dirAsc = ((t & k) == 0);
          if ((a < c) == dirAsc) { key[t] = c; key[ixj] = a; }  // descending
        }
      }
      __syncthreads();
    }
  for (int t = tid; t < KPAD; t += blockDim.x) {
    u64 kk = key[t];
    float s = unorderable((u32)(kk >> 32));
    u32 idx = ~((u32)kk);
    scoresk[b * KPAD + t] = s;
    float bx0 = 0, bx1 = 0, bx2 = 0, bx3 = 0; int lb = 0;
    if (idx < NCAND) {
      const float* bp = pboxes + ((size_t)b * NCAND + idx) * 4;
      bx0 = bp[0]; bx1 = bp[1]; bx2 = bp[2]; bx3 = bp[3];
      lb = plabels[(size_t)b * NCAND + idx];
    }
    float* ob = boxesk + ((size_t)b * KPAD + t) * 4;
    ob[0] = bx0; ob[1] = bx1; ob[2] = bx2; ob[3] = bx3;
    labelsk[b * KPAD + t] = lb;
  }
}

// -------------------------------------------- NMS (TDM tiles into LDS) ----
__global__ void __launch_bounds__(KPAD)
k_nms(const float* __restrict__ scoresk, const float* __restrict__ boxesk,
      const int* __restrict__ labelsk, float* __restrict__ out) {
  __shared__ float s_box[KPAD * 4];          // original boxes (TDM)
  __shared__ float s_score[KPAD];            // (TDM)
  __shared__ int   s_label[KPAD];            // (TDM)
  __shared__ float s_nms[KPAD * 4];          // label-offset boxes
  __shared__ float s_area[KPAD];
  __shared__ float s_red[KPAD];
  __shared__ u32   s_supp[KPAD][KPAD / 32];  // 32 KB suppression bitmask
  __shared__ u32   s_keep[KPAD / 32];
  __shared__ int   s_order[KEEP_MAX];
  __shared__ int   s_nkeep;

  const int b = blockIdx.x, tid = threadIdx.x;

  // wave 0 issues three async TDM descriptors, waits once
  if (tid < 32) {
    tdm_load_1d(s_box,   boxesk  + (size_t)b * KPAD * 4, KPAD * 4);
    tdm_load_1d(s_score, scoresk + (size_t)b * KPAD,     KPAD);
    tdm_load_1d(s_label, labelsk + (size_t)b * KPAD,     KPAD);
    tdm_wait0();
  }
  if (tid < KPAD / 32) s_keep[tid] = 0u;
  __syncthreads();

  const bool valid = (tid < TOPK) && (s_score[tid] > CONF_THR);
  float x1 = s_box[tid * 4 + 0], y1 = s_box[tid * 4 + 1];
  float x2 = s_box[tid * 4 + 2], y2 = s_box[tid * 4 + 3];
  s_red[tid] = valid ? fmaxf(fmaxf(x1, y1), fmaxf(x2, y2)) : 0.0f;
  __syncthreads();
  for (int s = KPAD / 2; s > 0; s >>= 1) {
    if (tid < s) s_red[tid] = fmaxf(s_red[tid], s_red[tid + s]);
    __syncthreads();
  }
  const float off = (float)s_label[tid] * (s_red[0] + 1.0f);
  const float nx1 = x1 + off, ny1 = y1 + off, nx2 = x2 + off, ny2 = y2 + off;
  s_nms[tid * 4 + 0] = nx1; s_nms[tid * 4 + 1] = ny1;
  s_nms[tid * 4 + 2] = nx2; s_nms[tid * 4 + 3] = ny2;
  s_area[tid] = (nx2 - nx1) * (ny2 - ny1);
  __syncthreads();

  // suppression bitmask row (thread tid owns candidate tid)
  {
    const float ai = s_area[tid];
    for (int w = 0; w < KPAD / 32; ++w) {
      u32 bits = 0u;
      for (int jb = 0; jb < 32; ++jb) {
        int j = w * 32 + jb;
        float lx = fmaxf(nx1, s_nms[j * 4 + 0]);
        float ly = fmaxf(ny1, s_nms[j * 4 + 1]);
        float rx = fminf(nx2, s_nms[j * 4 + 2]);
        float ry = fminf(ny2, s_nms[j * 4 + 3]);
        float iw = fmaxf(rx - lx, 0.0f), ih = fmaxf(ry - ly, 0.0f);
        float inter = iw * ih;
        float iou = inter / (ai + s_area[j] - inter);
        if (iou > NMS_THR) bits |= (1u << jb);
      }
      s_supp[tid][w] = bits;
    }
  }
  __syncthreads();

  // serial greedy scan (matches lax.scan semantics; rank<=100 applied after)
  if (tid == 0) {
    int nk = 0;
    for (int i = 0; i < TOPK; ++i) {
      if (!(s_score[i] > CONF_THR)) continue;
      bool sup = false;
      for (int w = 0; w < KPAD / 32; ++w)
        if (s_keep[w] & s_supp[i][w]) { sup = true; break; }
      if (!sup) {
        s_keep[i >> 5] |= (1u << (i & 31));
        if (nk < KEEP_MAX) s_order[nk] = i;
        ++nk;
      }
    }
    s_nkeep = nk < KEEP_MAX ? nk : KEEP_MAX;
  }
  __syncthreads();

  if (tid < KEEP_MAX) {
    const int BK = BATCH * KEEP_MAX;
    float* o_ids = out;
    float* o_box = out + BK;
    float* o_lab = out + BK * 5;
    float* o_sco = out + BK * 6;
    float* o_val = out + BK * 7;
    const int o = b * KEEP_MAX + tid;
    if (tid < s_nkeep) {
      int k = s_order[tid];
      o_ids[o] = (float)b;
      o_box[o * 4 + 0] = s_box[k * 4 + 0];
      o_box[o * 4 + 1] = s_box[k * 4 + 1];
      o_box[o * 4 + 2] = s_box[k * 4 + 2];
      o_box[o * 4 + 3] = s_box[k * 4 + 3];
      o_lab[o] = (float)s_label[k];
      o_sco[o] = s_score[k];
      o_val[o] = 1.0f;
    } else {
      o_ids[o] = -1.0f;
      o_box[o * 4 + 0] = 0.0f; o_box[o * 4 + 1] = 0.0f;
      o_box[o * 4 + 2] = 0.0f; o_box[o * 4 + 3] = 0.0f;
      o_lab[o] = -1.0f;
      o_sco[o] = 0.0f;
      o_val[o] = 0.0f;
    }
  }
}

// ------------------------------------------------------------- launch ----
extern "C" void kernel_launch(void* const* d_in, const int* in_sizes, int n_in,
                              void* d_out, int out_size, void* d_ws, size_t ws_size,
                              hipStream_t stream) {
  const float* pscores = (const float*)d_in[0];
  const float* pboxes  = (const float*)d_in[1];
  const int*   plabels = (const int*)d_in[2];
  float* out = (float*)d_out;

  u32*   g_hist = (u32*)d_ws;                        // BATCH*NBINS
  u32*   g_cut  = g_hist + BATCH * NBINS;            // BATCH*4
  float* g_csc  = (float*)(g_cut + BATCH * 4);       // BATCH*KPAD
  u32*   g_cix  = (u32*)(g_csc + BATCH * KPAD);      // BATCH*KPAD
  float* g_sco  = (float*)(g_cix + BATCH * KPAD);    // BATCH*KPAD
  float* g_box  = g_sco + BATCH * KPAD;              // BATCH*KPAD*4
  int*   g_lab  = (int*)(g_box + BATCH * KPAD * 4);  // BATCH*KPAD
  u32*   g_bhi  = (u32*)(g_lab + BATCH * KPAD);      // BATCH*SPLIT
  u32*   g_beq  = g_bhi + BATCH * SPLIT;             // BATCH*SPLIT
  u32*   g_ohi  = g_beq + BATCH * SPLIT;             // BATCH*SPLIT
  u32*   g_oeq  = g_ohi + BATCH * SPLIT;             // BATCH*SPLIT

  k_init <<<dim3(240),          dim3(256), 0, stream>>>(g_hist, g_csc, g_cix);
  k_hist <<<dim3(BATCH, SPLIT), dim3(256), 0, stream>>>(pscores, g_hist);
  k_cut  <<<dim3(1),            dim3(32),  0, stream>>>(g_hist, g_cut);
  k_cnt  <<<dim3(BATCH, SPLIT), dim3(256), 0, stream>>>(pscores, g_cut, g_bhi, g_beq);
  k_scan <<<dim3(1),            dim3(32),  0, stream>>>(g_bhi, g_beq, g_ohi, g_oeq);
  k_place<<<dim3(BATCH, SPLIT), dim3(256), 0, stream>>>(pscores, g_cut, g_ohi, g_oeq,
                                                        g_csc, g_cix);
  k_sort <<<dim3(BATCH),        dim3(256), 0, stream>>>(pboxes, plabels, g_csc, g_cix,
                                                        g_sco, g_box, g_lab);
  k_nms  <<<dim3(BATCH),        dim3(KPAD), 0, stream>>>(g_sco, g_box, g_lab, out);
}